// SumLayer_4028679323772
// MI455X (gfx1250) — compile-verified
//
#include <hip/hip_runtime.h>

#define EDGES 32
#define BATCH 128

// ---- gfx1250 async global->LDS path detection -------------------------------
#if defined(__has_builtin)
#  if __has_builtin(__builtin_amdgcn_global_load_async_to_lds_b32)
#    define HAVE_ASYNC_B32 1
#  endif
#  if __has_builtin(__builtin_amdgcn_s_wait_asynccnt)
#    define HAVE_WAIT_ASYNC 1
#  endif
#endif

#if HAVE_WAIT_ASYNC
#  define WAIT_ASYNC0() __builtin_amdgcn_s_wait_asynccnt(0)
#else
#  define WAIT_ASYNC0() asm volatile("s_wait_asynccnt 0" ::: "memory")
#endif

typedef __attribute__((address_space(1))) int* g32_t;   // global int*
typedef __attribute__((address_space(3))) int* l32_t;   // LDS int*

// One block (128 threads = 4 wave32) per sum node. Thread t owns batch col t.
//  * cids/pids/params indexing is block-uniform -> SMEM s_loads, SGPR weights.
//  * The 16KB child tile is gathered with gfx1250 async global->LDS DMA; the
//    per-edge address is a 32-bit byte offset (cid<<9, max 2^27) so the scalar
//    part is one s_lshl_b32 and the lane part one u32 add (GVS addressing).
//  * Each lane's async ops write exactly the LDS words the same lane later
//    reads, so a per-wave s_wait_asynccnt is sufficient -- no barriers.
__global__ __launch_bounds__(BATCH) void sum_layer_fwd(
    const float* __restrict__ element_mars,   // [MAX_NUM_ELS, BATCH]
    const float* __restrict__ params,         // [NUM_PARAMS]
    const int*   __restrict__ nids,           // [NUM_NODES]
    const int*   __restrict__ cids,           // [NUM_NODES, EDGES]
    const int*   __restrict__ pids,           // [NUM_NODES, EDGES]
    float*       __restrict__ out)            // [NUM_NODES, BATCH]
{
    __shared__ alignas(16) float s_ch[EDGES * BATCH];   // 16 KB gathered rows

    const int n = blockIdx.x;                 // uniform
    const int t = threadIdx.x;                // batch column

    // ---- Stage 1: async-gather 32 child rows (512B each, coalesced) --------
#if HAVE_ASYNC_B32
    #pragma unroll
    for (int e = 0; e < EDGES; ++e) {
        const int cid = cids[n * EDGES + e];               // uniform s_load
        const unsigned boff = ((unsigned)cid << 9) + ((unsigned)t << 2);
        const char* g = (const char*)element_mars + boff;
        float*      l = &s_ch[e * BATCH + t];
        __builtin_amdgcn_global_load_async_to_lds_b32((g32_t)g, (l32_t)l,
                                                      /*offset=*/0, /*cpol=*/0);
    }
#else
    #pragma unroll
    for (int e = 0; e < EDGES; ++e) {
        const int cid = cids[n * EDGES + e];
        s_ch[e * BATCH + t] = element_mars[(size_t)(unsigned)cid * BATCH + t];
    }
#endif

    // Uniform weight gather overlaps the DMA: chained scalar loads -> SGPRs.
    float w[EDGES];
    #pragma unroll
    for (int e = 0; e < EDGES; ++e)
        w[e] = params[pids[n * EDGES + e]];                // s_load chain

#if HAVE_ASYNC_B32
    WAIT_ASYNC0();    // per-wave drain; lane-local LDS visibility is enough
#endif

    // ---- Stage 2: two-pass log-sum-exp over 32 edges ------------------------
    float m = -3.402823466e38f;
    #pragma unroll
    for (int e = 0; e < EDGES; ++e)
        m = fmaxf(m, s_ch[e * BATCH + t]);

    float s = 0.0f;
    #pragma unroll
    for (int e = 0; e < EDGES; ++e)
        s = __builtin_fmaf(w[e], __expf(s_ch[e * BATCH + t] - m), s);

    const float v = __logf(fmaxf(s, 1e-10f)) + m;

    // ---- Stage 3: scatter the result row (nids[n] uniform -> s_load) --------
    out[(size_t)(unsigned)nids[n] * BATCH + t] = v;
}

extern "C" void kernel_launch(void* const* d_in, const int* in_sizes, int n_in,
                              void* d_out, int out_size, void* d_ws, size_t ws_size,
                              hipStream_t stream) {
    const float* node_mars    = (const float*)d_in[0];
    const float* element_mars = (const float*)d_in[1];
    const float* params       = (const float*)d_in[2];
    const int*   nids         = (const int*)d_in[3];
    const int*   cids         = (const int*)d_in[4];
    const int*   pids         = (const int*)d_in[5];
    float*       out          = (float*)d_out;

    const int num_nodes = in_sizes[3];   // |nids|

    // General scatter semantics: rows not named by nids keep node_mars values.
    (void)hipMemcpyAsync(out, node_mars, (size_t)out_size * sizeof(float),
                         hipMemcpyDeviceToDevice, stream);

    sum_layer_fwd<<<num_nodes, BATCH, 0, stream>>>(element_mars, params,
                                                   nids, cids, pids, out);
}